// MP_34686155882688
// MI455X (gfx1250) — compile-verified
//
#include <hip/hip_runtime.h>

typedef __attribute__((ext_vector_type(2))) float v2f;
typedef __attribute__((ext_vector_type(8))) float v8f;

#define D_DIM 128   // output/feature dim of every layer in this model

// ---------------------------------------------------------------------------
// Fused WMMA GEMM:  out[N, 128] = act( A[N, K] @ W[K, 128] + bias )
// A is the (virtual) concatenation of A0[N, K0] and A1[N, K-K0].
// One wave32 computes one 16x16 output tile using V_WMMA_F32_16X16X4_F32,
// marching K in steps of 4 (full fp32 matrix pipe, no precision loss).
// ---------------------------------------------------------------------------
__global__ __launch_bounds__(128) void wmma_gemm_f32_kernel(
    const float* __restrict__ A0, const float* __restrict__ A1,
    int K0, int K, const float* __restrict__ W,
    const float* __restrict__ bias, float* __restrict__ out,
    int N, int do_relu)
{
    const int lane   = threadIdx.x;                      // 0..31
    const int tile_m = blockIdx.x * blockDim.y + threadIdx.y;
    const int tile_n = blockIdx.y;
    if (tile_m >= (N >> 4)) return;                      // wave-uniform guard

    const int row0 = tile_m << 4;
    const int n0   = tile_n << 4;

    // A-matrix (16x4 f32) lane mapping: row = lane&15, K-pair = (lane>>4)*2
    const int am = lane & 15;
    const int kp = (lane >> 4) << 1;                     // 0 or 2
    // B-matrix (4x16 f32) lane mapping mirrors A: col = lane&15, same K-pair
    const int bn = lane & 15;

    v8f c = {};                                          // 16x16 f32 accum (8 VGPRs)

    for (int k0 = 0; k0 < K; k0 += 4) {
        const int ka = k0 + kp;                          // this lane's first K index

        // ---- load A fragment: 2 contiguous K values for (row0+am) ----
        v2f a;
        if (ka < K0) {
            const float* p = A0 + (size_t)(row0 + am) * K0 + ka;
            a.x = p[0]; a.y = p[1];
        } else {
            const float* p = A1 + (size_t)(row0 + am) * (K - K0) + (ka - K0);
            a.x = p[0]; a.y = p[1];
        }

        // ---- load B fragment: W[ka][n0+bn], W[ka+1][n0+bn] (stride 128) ----
        v2f b;
        b.x = W[(size_t)ka * D_DIM + n0 + bn];
        b.y = W[(size_t)(ka + 1) * D_DIM + n0 + bn];

        // D = A x B + C   (v_wmma_f32_16x16x4_f32)
        c = __builtin_amdgcn_wmma_f32_16x16x4_f32(
                /*neg_a=*/false, a, /*neg_b=*/false, b,
                /*c_mod=*/(short)0, c, /*reuse_a=*/false, /*reuse_b=*/false);
    }

    // C/D layout: VGPR r -> row (r + 8*(lane>>4)), col lane&15
    const float bv    = bias[n0 + (lane & 15)];
    const int   mbase = (lane >> 4) << 3;
    const int   col   = n0 + (lane & 15);
#pragma unroll
    for (int r = 0; r < 8; ++r) {
        float v = c[r] + bv;
        if (do_relu) v = fmaxf(v, 0.0f);
        out[(size_t)(row0 + mbase + r) * D_DIM + col] = v;
    }
}

// ---------------------------------------------------------------------------
// Edge scatter: z[dst[e], :] += y[src[e], :], indeg[dst[e]] += 1
// One block of 128 threads per edge iteration (1 float per thread),
// grid-stride over E. Atomic f32 adds (global_atomic_add_f32).
// ---------------------------------------------------------------------------
__global__ __launch_bounds__(128) void scatter_edges_kernel(
    const float* __restrict__ y, const int* __restrict__ src,
    const int* __restrict__ dst, float* __restrict__ z,
    float* __restrict__ indeg, int E)
{
    for (int e = blockIdx.x; e < E; e += gridDim.x) {
        const int s = src[e];
        const int d = dst[e];
        const float v = y[(size_t)s * D_DIM + threadIdx.x];
        atomicAdd(&z[(size_t)d * D_DIM + threadIdx.x], v);
        if (threadIdx.x == 0) atomicAdd(&indeg[d], 1.0f);
    }
}

// ---------------------------------------------------------------------------
// zsel[n, d] = indeg[n] > 0 ? z[n, d] : z_init[n, d]
// ---------------------------------------------------------------------------
__global__ __launch_bounds__(256) void select_z_kernel(
    const float* __restrict__ z, const float* __restrict__ z_init,
    const float* __restrict__ indeg, float* __restrict__ zsel, int total)
{
    int i = blockIdx.x * blockDim.x + threadIdx.x;
    if (i < total) {
        const int row = i >> 7;                          // / 128
        zsel[i] = (indeg[row] > 0.0f) ? z[i] : z_init[i];
    }
}

extern "C" void kernel_launch(void* const* d_in, const int* in_sizes, int n_in,
                              void* d_out, int out_size, void* d_ws, size_t ws_size,
                              hipStream_t stream) {
    const float* x      = (const float*)d_in[0];
    const float* z_init = (const float*)d_in[1];
    const float* W_pre  = (const float*)d_in[2];
    const float* b_pre  = (const float*)d_in[3];
    const float* W_u1   = (const float*)d_in[4];
    const float* b_u1   = (const float*)d_in[5];
    const float* W_u2   = (const float*)d_in[6];
    const float* b_u2   = (const float*)d_in[7];
    const int*   src    = (const int*)d_in[8];
    const int*   dst    = (const int*)d_in[9];
    float*       out    = (float*)d_out;

    const int N = in_sizes[0] / D_DIM;     // 40000
    const int E = in_sizes[8];             // 640000
    const size_t ND = (size_t)N * D_DIM;

    // workspace layout (floats)
    float* ws    = (float*)d_ws;
    float* y     = ws;                     // relu(x @ W_pre + b_pre)   [N,128]
    float* z     = y + ND;                 // segment-sum accumulator   [N,128]
    float* zsel  = z + ND;                 // z after no-in-edge select [N,128]
    float* t     = zsel + ND;              // hidden after first update [N,128]
    float* indeg = t + ND;                 // in-degree (as float)      [N]

    hipMemsetAsync(z,     0, ND * sizeof(float), stream);
    hipMemsetAsync(indeg, 0, (size_t)N * sizeof(float), stream);

    const dim3 blk(32, 4);                               // 4 waves / block
    const int tiles_m = N / 16;                          // 2500
    const dim3 grid((tiles_m + 3) / 4, D_DIM / 16);      // 625 x 8

    // 1) per-node message precompute: y = relu(x @ W_pre + b_pre), K=128
    wmma_gemm_f32_kernel<<<grid, blk, 0, stream>>>(x, x, 128, 128, W_pre, b_pre, y, N, 1);

    // 2) mailbox scatter-add + in-degree
    scatter_edges_kernel<<<dim3(4096), dim3(128), 0, stream>>>(y, src, dst, z, indeg, E);

    // 3) keep z_init where no in-edges
    const int total = (int)ND;
    select_z_kernel<<<(total + 255) / 256, 256, 0, stream>>>(z, z_init, indeg, zsel, total);

    // 4) t = relu(concat(x, zsel) @ W_u1 + b_u1), K=256 split 128|128
    wmma_gemm_f32_kernel<<<grid, blk, 0, stream>>>(x, zsel, 128, 256, W_u1, b_u1, t, N, 1);

    // 5) out = t @ W_u2 + b_u2, K=128, no relu
    wmma_gemm_f32_kernel<<<grid, blk, 0, stream>>>(t, t, 128, 128, W_u2, b_u2, out, N, 0);
}